// CodeGraphSAGE_69286412419258
// MI455X (gfx1250) — compile-verified
//
#include <hip/hip_runtime.h>

typedef __attribute__((ext_vector_type(2))) float v2f;
typedef __attribute__((ext_vector_type(8))) float v8f;

#define IN_C  8
#define HID_C 64
#define OUT_C 32

// ---------------------------------------------------------------------------
// Edge kernel, layer 1: degree count + sum of x[src] into agg1[dst].
// One thread per edge; 8-channel rows loaded as two float4.
// ---------------------------------------------------------------------------
__global__ void sage_edge_agg1(const float* __restrict__ x,
                               const int* __restrict__ src,
                               const int* __restrict__ dst,
                               float* __restrict__ cnt,
                               float* __restrict__ agg1,
                               int n_edges) {
  int e = blockIdx.x * blockDim.x + threadIdx.x;
  if (e >= n_edges) return;
  int s = src[e];
  int d = dst[e];
  atomicAdd(&cnt[d], 1.0f);
  const float4* xr = (const float4*)(x + (size_t)s * IN_C);
  float4 a = xr[0];
  float4 b = xr[1];
  float vals[8];
  *(float4*)(vals + 0) = a;
  *(float4*)(vals + 4) = b;
  float* o = agg1 + (size_t)d * IN_C;
#pragma unroll
  for (int c = 0; c < 8; ++c) atomicAdd(o + c, vals[c]);
}

// ---------------------------------------------------------------------------
// Edge kernel, layer 2: sum of h[src] (64ch) into agg2[dst].
// Four threads per edge, 16 channels each (four float4 loads).
// ---------------------------------------------------------------------------
__global__ void sage_edge_agg2(const float* __restrict__ h,
                               const int* __restrict__ src,
                               const int* __restrict__ dst,
                               float* __restrict__ agg2,
                               int n_edges) {
  long long t = (long long)blockIdx.x * blockDim.x + threadIdx.x;
  int e = (int)(t >> 2);
  int q = (int)(t & 3);
  if (e >= n_edges) return;
  int s = src[e];
  int d = dst[e];
  const float4* hr = (const float4*)(h + (size_t)s * HID_C + q * 16);
  float vals[16];
#pragma unroll
  for (int i = 0; i < 4; ++i) *(float4*)(vals + 4 * i) = hr[i];
  float* o = agg2 + (size_t)d * HID_C + q * 16;
#pragma unroll
  for (int c = 0; c < 16; ++c) atomicAdd(o + c, vals[c]);
}

// ---------------------------------------------------------------------------
// Layer 1 projection: h = relu(mean1 @ W1l^T + b1 + x @ W1r^T)
// Transposed mapping: D[channel][node] = W (A operand) x feat^T (B operand).
// One wave per 16-node tile; 64 out channels -> 4 n-tiles; K=8 -> 2 k-steps.
// Epilogue: each lane owns 8 consecutive channels of one node -> b128 stores.
// ---------------------------------------------------------------------------
__global__ void __launch_bounds__(256) sage_l1_wmma(
    const float* __restrict__ x,
    const float* __restrict__ W1l,   // [64,8] row-major (out, in)
    const float* __restrict__ W1r,   // [64,8]
    const float* __restrict__ b1,    // [64]
    const float* __restrict__ cnt,
    const float* __restrict__ agg1,  // [N,8]
    float* __restrict__ h,           // [N,64]
    int n_nodes, int n_tiles) {
  __shared__ float lwl[HID_C * IN_C];
  __shared__ float lwr[HID_C * IN_C];
  __shared__ float lb[HID_C];
  for (int i = threadIdx.x; i < HID_C * IN_C; i += 256) {
    lwl[i] = W1l[i];
    lwr[i] = W1r[i];
  }
  if (threadIdx.x < HID_C) lb[threadIdx.x] = b1[threadIdx.x];
  __syncthreads();

  int wave  = threadIdx.x >> 5;
  int lane  = threadIdx.x & 31;
  int tile  = blockIdx.x * 8 + wave;
  if (tile >= n_tiles) return;           // wave-uniform: EXEC stays all-ones

  int row   = lane & 15;                 // channel row (A) / node col (B)
  int khalf = lane >> 4;                 // lanes 16..31 carry K+2 (A/B layout)
  int node  = tile * 16 + row;
  int nodec = node < n_nodes ? node : (n_nodes - 1);

  float inv = 1.0f / fmaxf(cnt[nodec], 1.0f);
  const v2f* arow = (const v2f*)(agg1 + (size_t)nodec * IN_C);
  const v2f* xrow = (const v2f*)(x    + (size_t)nodec * IN_C);
  const v2f* wlp  = (const v2f*)lwl;
  const v2f* wrp  = (const v2f*)lwr;

  v8f acc[4] = {};
#pragma unroll
  for (int kk = 0; kk < 2; ++kk) {       // K = kk*4 .. kk*4+3
    v2f bm = arow[kk * 2 + khalf];
    bm = bm * inv;                        // mean = agg / max(cnt,1)
    v2f bx = xrow[kk * 2 + khalf];
#pragma unroll
    for (int nb = 0; nb < 4; ++nb) {
      v2f al = wlp[(nb * 16 + row) * 4 + kk * 2 + khalf];
      v2f ar = wrp[(nb * 16 + row) * 4 + kk * 2 + khalf];
      acc[nb] = __builtin_amdgcn_wmma_f32_16x16x4_f32(
          false, al, false, bm, (short)0, acc[nb], false, false);
      acc[nb] = __builtin_amdgcn_wmma_f32_16x16x4_f32(
          false, ar, false, bx, (short)0, acc[nb], false, false);
    }
  }

  // D layout: VGPR r -> channel nb*16 + khalf*8 + r, column N = node (per lane)
  if (node < n_nodes) {
    float* orow = h + (size_t)node * HID_C + khalf * 8;
#pragma unroll
    for (int nb = 0; nb < 4; ++nb) {
      const float4* bv = (const float4*)(lb + nb * 16 + khalf * 8);
      float4 b0 = bv[0], b1v = bv[1];
      float4 lo, hi;
      lo.x = fmaxf(acc[nb][0] + b0.x, 0.0f);
      lo.y = fmaxf(acc[nb][1] + b0.y, 0.0f);
      lo.z = fmaxf(acc[nb][2] + b0.z, 0.0f);
      lo.w = fmaxf(acc[nb][3] + b0.w, 0.0f);
      hi.x = fmaxf(acc[nb][4] + b1v.x, 0.0f);
      hi.y = fmaxf(acc[nb][5] + b1v.y, 0.0f);
      hi.z = fmaxf(acc[nb][6] + b1v.z, 0.0f);
      hi.w = fmaxf(acc[nb][7] + b1v.w, 0.0f);
      *(float4*)(orow + nb * 16)     = lo;
      *(float4*)(orow + nb * 16 + 4) = hi;
    }
  }
}

// ---------------------------------------------------------------------------
// Layer 2 projection: out = mean2 @ W2l^T + b2 + h @ W2r^T   (transposed tile)
// One wave per 16-node tile; 32 out channels -> 2 n-tiles; K=64 -> 16 k-steps.
// ---------------------------------------------------------------------------
__global__ void __launch_bounds__(256) sage_l2_wmma(
    const float* __restrict__ W2l,   // [32,64]
    const float* __restrict__ W2r,   // [32,64]
    const float* __restrict__ b2,    // [32]
    const float* __restrict__ cnt,
    const float* __restrict__ agg2,  // [N,64]
    const float* __restrict__ h,     // [N,64]
    float* __restrict__ out,         // [N,32]
    int n_nodes, int n_tiles) {
  __shared__ float lwl[OUT_C * HID_C];
  __shared__ float lwr[OUT_C * HID_C];
  __shared__ float lb[OUT_C];
  for (int i = threadIdx.x; i < OUT_C * HID_C; i += 256) {
    lwl[i] = W2l[i];
    lwr[i] = W2r[i];
  }
  if (threadIdx.x < OUT_C) lb[threadIdx.x] = b2[threadIdx.x];
  __syncthreads();

  int wave  = threadIdx.x >> 5;
  int lane  = threadIdx.x & 31;
  int tile  = blockIdx.x * 8 + wave;
  if (tile >= n_tiles) return;

  int row   = lane & 15;
  int khalf = lane >> 4;
  int node  = tile * 16 + row;
  int nodec = node < n_nodes ? node : (n_nodes - 1);

  float inv = 1.0f / fmaxf(cnt[nodec], 1.0f);
  const v2f* arow = (const v2f*)(agg2 + (size_t)nodec * HID_C);
  const v2f* hrow = (const v2f*)(h    + (size_t)nodec * HID_C);
  const v2f* wlp  = (const v2f*)lwl;
  const v2f* wrp  = (const v2f*)lwr;

  v8f acc[2] = {};
#pragma unroll
  for (int kk = 0; kk < 16; ++kk) {      // K = kk*4 .. kk*4+3
    v2f bm = arow[kk * 2 + khalf];
    bm = bm * inv;
    v2f bh = hrow[kk * 2 + khalf];
#pragma unroll
    for (int nb = 0; nb < 2; ++nb) {
      v2f al = wlp[(nb * 16 + row) * 32 + kk * 2 + khalf];
      v2f ar = wrp[(nb * 16 + row) * 32 + kk * 2 + khalf];
      acc[nb] = __builtin_amdgcn_wmma_f32_16x16x4_f32(
          false, al, false, bm, (short)0, acc[nb], false, false);
      acc[nb] = __builtin_amdgcn_wmma_f32_16x16x4_f32(
          false, ar, false, bh, (short)0, acc[nb], false, false);
    }
  }

  if (node < n_nodes) {
    float* orow = out + (size_t)node * OUT_C + khalf * 8;
#pragma unroll
    for (int nb = 0; nb < 2; ++nb) {
      const float4* bv = (const float4*)(lb + nb * 16 + khalf * 8);
      float4 b0 = bv[0], b1v = bv[1];
      float4 lo, hi;
      lo.x = acc[nb][0] + b0.x;
      lo.y = acc[nb][1] + b0.y;
      lo.z = acc[nb][2] + b0.z;
      lo.w = acc[nb][3] + b0.w;
      hi.x = acc[nb][4] + b1v.x;
      hi.y = acc[nb][5] + b1v.y;
      hi.z = acc[nb][6] + b1v.z;
      hi.w = acc[nb][7] + b1v.w;
      *(float4*)(orow + nb * 16)     = lo;
      *(float4*)(orow + nb * 16 + 4) = hi;
    }
  }
}

// ---------------------------------------------------------------------------
extern "C" void kernel_launch(void* const* d_in, const int* in_sizes, int n_in,
                              void* d_out, int out_size, void* d_ws, size_t ws_size,
                              hipStream_t stream) {
  const float* x   = (const float*)d_in[0];
  const int*   ei  = (const int*)  d_in[1];
  const float* W1l = (const float*)d_in[2];
  const float* W1r = (const float*)d_in[3];
  const float* b1  = (const float*)d_in[4];
  const float* W2l = (const float*)d_in[5];
  const float* W2r = (const float*)d_in[6];
  const float* b2  = (const float*)d_in[7];
  float* out = (float*)d_out;

  int N = in_sizes[0] / IN_C;
  int E = in_sizes[1] / 2;
  const int* src = ei;
  const int* dst = ei + E;

  // workspace layout (floats): cnt[N] | pad | agg1[N*8] | h[N*64] | agg2[N*64]
  float* ws = (float*)d_ws;
  size_t npad     = ((size_t)N + 63) & ~(size_t)63;
  size_t agg1_off = npad;
  size_t h_off    = agg1_off + (size_t)N * IN_C;
  size_t agg2_off = h_off + (size_t)N * HID_C;
  float* cnt  = ws;
  float* agg1 = ws + agg1_off;
  float* hbuf = ws + h_off;
  float* agg2 = ws + agg2_off;

  // zero the accumulators (cnt..agg1 contiguous, then agg2)
  hipMemsetAsync(ws, 0, (agg1_off + (size_t)N * IN_C) * sizeof(float), stream);
  hipMemsetAsync(agg2, 0, (size_t)N * HID_C * sizeof(float), stream);

  const int blk = 256;
  int tiles = (N + 15) / 16;

  sage_edge_agg1<<<(E + blk - 1) / blk, blk, 0, stream>>>(x, src, dst, cnt, agg1, E);
  sage_l1_wmma<<<(tiles + 7) / 8, 256, 0, stream>>>(x, W1l, W1r, b1, cnt, agg1,
                                                    hbuf, N, tiles);
  long long t2 = (long long)E * 4;
  sage_edge_agg2<<<(int)((t2 + blk - 1) / blk), blk, 0, stream>>>(hbuf, src, dst,
                                                                  agg2, E);
  sage_l2_wmma<<<(tiles + 7) / 8, 256, 0, stream>>>(W2l, W2r, b2, cnt, agg2,
                                                    hbuf, out, N, tiles);
}